// CLoss_49924699848800
// MI455X (gfx1250) — compile-verified
//
#include <hip/hip_runtime.h>
#include <hip/hip_bf16.h>
#include <math.h>

typedef __attribute__((ext_vector_type(2))) float v2f;
typedef __attribute__((ext_vector_type(8))) float v8f;

#define NROWS 32768
#define NCOLS 1000
#define NVEC  250      // 1000 / 4

// Full 32-lane f32 sum via V_WMMA_F32_16X16X4_F32 with all-ones B.
// A layout (16x4 f32): lanes 0-15 hold {K0,K1} in {v0,v1}, lanes 16-31 hold {K2,K3}.
// With a = {x, 0} and B = ones, D[m][n] = x_m + x_{m+16} for every n.
// C/D layout: lane n<16 vgpr j = D[j][n]; lane n>=16 vgpr j = D[j+8][n-16].
// Sum the 8 C vgprs + shfl_xor(16) => every lane holds the full 32-lane sum.
__device__ __forceinline__ float wave_sum_wmma(float x) {
  v2f a; a[0] = x;    a[1] = 0.0f;
  v2f b; b[0] = 1.0f; b[1] = 1.0f;
  v8f c = {0.f, 0.f, 0.f, 0.f, 0.f, 0.f, 0.f, 0.f};
  c = __builtin_amdgcn_wmma_f32_16x16x4_f32(false, a, false, b, (short)0, c,
                                            false, false);
  float s = ((c[0] + c[1]) + (c[2] + c[3])) + ((c[4] + c[5]) + (c[6] + c[7]));
  s += __shfl_xor(s, 16, 32);
  return s;
}

// Phase 1: one wave per row. Online (flash-style) max/sum-exp, single pass.
// Memory-bound: 131 MB streamed once, float4 coalesced loads.
__global__ __launch_bounds__(256)
void closs_rows_kernel(const float* __restrict__ out,
                       const int* __restrict__ tgt,
                       float* __restrict__ lbuf,
                       unsigned int* __restrict__ negcnt) {
  __shared__ unsigned int blkNeg;
  if (threadIdx.x == 0) blkNeg = 0u;
  __syncthreads();

  const int lane = threadIdx.x & 31;
  const int wid  = threadIdx.x >> 5;
  const int row  = blockIdx.x * 8 + wid;

  const float*  rowp = out + (size_t)row * NCOLS;
  const float4* rowv = reinterpret_cast<const float4*>(rowp);

  float m = -INFINITY;
  float s = 0.0f;
#pragma unroll
  for (int i = 0; i < 8; ++i) {
    int idx = lane + 32 * i;
    if (idx < NVEC) {
      float4 v  = rowv[idx];
      float  vm = fmaxf(fmaxf(v.x, v.y), fmaxf(v.z, v.w));
      float  mo = m;
      m = fmaxf(m, vm);
      s = s * __expf(mo - m) + __expf(v.x - m) + __expf(v.y - m) +
          __expf(v.z - m) + __expf(v.w - m);
    }
  }

  // Wave-wide max (shuffle tree), then WMMA-based sum of rescaled partials.
  float M = m;
#pragma unroll
  for (int off = 16; off >= 1; off >>= 1)
    M = fmaxf(M, __shfl_xor(M, off, 32));
  float sadj = s * __expf(m - M);
  float S = wave_sum_wmma(sadj);   // uniform control flow: EXEC all ones

  int   t   = tgt[row];
  float zt  = rowp[t];             // broadcast load (one fetch per wave)
  float lse = M + __logf(S);
  // margin == 0  <=>  z_t == row max  => loss = relu(1 - 0) = 1
  float loss = (zt == M) ? 1.0f : fmaxf(0.0f, 1.0f - zt + lse);

  if (lane == 0) {
    lbuf[row] = loss;
    if (zt != M) atomicAdd(&blkNeg, 1u);   // margin < 0 count
  }
  __syncthreads();
  if (threadIdx.x == 0 && blkNeg) atomicAdd(negcnt, blkNeg);
}

// Phase 2: single 1024-thread block; all 32768 losses live in registers
// (32 per thread). Selection mask is provably a prefix of the sorted order
// (losses >= 0 => cum[i]+i strictly increasing), so binary-search the float
// bit space for the largest fully-selected value, then resolve the tie group.
__global__ __launch_bounds__(1024)
void closs_select_kernel(const float* __restrict__ lbuf,
                         const unsigned int* __restrict__ negcnt,
                         float* __restrict__ outv) {
  __shared__ float redA[32];
  __shared__ float redB[32];

  const int tid  = threadIdx.x;
  const int lane = tid & 31;
  const int wid  = tid >> 5;

  float vals[32];
#pragma unroll
  for (int i = 0; i < 32; ++i) vals[i] = lbuf[i * 1024 + tid];

  auto block_sum2 = [&](float x, float y, float& ox, float& oy) {
    float wx = wave_sum_wmma(x);
    float wy = wave_sum_wmma(y);
    __syncthreads();                       // protect redA/redB reuse
    if (lane == 0) { redA[wid] = wx; redB[wid] = wy; }
    __syncthreads();
    ox = wave_sum_wmma(redA[lane]);        // 32 waves -> every thread gets total
    oy = wave_sum_wmma(redB[lane]);
  };

  const float THR = 32769.0f;              // B + 1
  unsigned int cur = 0u;
  for (int b = 30; b >= 0; --b) {
    unsigned int trial = cur | (1u << b);
    if (trial > 0x7F800000u) continue;     // stay below +inf (uniform)
    float t = __uint_as_float(trial);
    float c = 0.0f, s = 0.0f;
#pragma unroll
    for (int i = 0; i < 32; ++i)
      if (vals[i] <= t) { c += 1.0f; s += vals[i]; }
    float C, S;
    block_sum2(c, s, C, S);
    if (S + C <= THR) cur = trial;         // same totals everywhere -> uniform
  }

  // Stats at the found cut value.
  float tq = __uint_as_float(cur);
  float c = 0.0f, s = 0.0f;
#pragma unroll
  for (int i = 0; i < 32; ++i)
    if (vals[i] <= tq) { c += 1.0f; s += vals[i]; }
  float C0, S0;
  block_sum2(c, s, C0, S0);
  bool  ok   = (S0 + C0 <= THR);           // false only if even t=0 fails
  float tcut = ok ? tq : -1.0f;
  float n0   = ok ? C0 : 0.0f;
  float s0   = ok ? S0 : 0.0f;

  // Smallest value strictly above the cut (boundary/tie value).
  float mn = INFINITY;
#pragma unroll
  for (int i = 0; i < 32; ++i)
    if (vals[i] > tcut) mn = fminf(mn, vals[i]);
#pragma unroll
  for (int off = 16; off >= 1; off >>= 1)
    mn = fminf(mn, __shfl_xor(mn, off, 32));
  __syncthreads();
  if (lane == 0) redA[wid] = mn;
  __syncthreads();
  float v = redA[lane];
#pragma unroll
  for (int off = 16; off >= 1; off >>= 1)
    v = fminf(v, __shfl_xor(v, off, 32));

  if (tid == 0) {
    float R = THR - s0 - n0;
    float j = 0.0f, add = 0.0f;
    if (v < INFINITY) {                    // partial selection inside tie group
      j = floorf(R / (v + 1.0f));
      j = fmaxf(j, 0.0f);
      j = fminf(j, 32768.0f - n0);
      add = j * v;
    }
    float cl1 = s0 + add;                              // dot(v, l)
    float cl2 = 32768.0f - (n0 + j) + (float)(*negcnt); // B - sum(v) + #(margin<0)
    outv[0] = (cl1 < cl2) ? cl2 : cl1;
  }
}

extern "C" void kernel_launch(void* const* d_in, const int* in_sizes, int n_in,
                              void* d_out, int out_size, void* d_ws, size_t ws_size,
                              hipStream_t stream) {
  (void)in_sizes; (void)n_in; (void)out_size; (void)ws_size;
  const float* out = (const float*)d_in[0];
  const int*   tgt = (const int*)d_in[1];

  float*        lbuf = (float*)d_ws;                       // 32768 floats
  unsigned int* neg  = (unsigned int*)((char*)d_ws + NROWS * sizeof(float));

  hipMemsetAsync(neg, 0, sizeof(unsigned int), stream);
  closs_rows_kernel<<<NROWS / 8, 256, 0, stream>>>(out, tgt, lbuf, neg);
  closs_select_kernel<<<1, 1024, 0, stream>>>(lbuf, neg, (float*)d_out);
}